// NALU_18554258719020
// MI455X (gfx1250) — compile-verified
//
#include <hip/hip_runtime.h>
#include <math.h>

typedef float v2f __attribute__((ext_vector_type(2)));
typedef float v4f __attribute__((ext_vector_type(4)));
typedef float v8f __attribute__((ext_vector_type(8)));

#define NB 1024
#define NI 512
#define NO 512
#define EPS 1e-7f
#define LOG_CLIP 20.0f
#define NEG_BIG -1e30f

// PA record (16 B) per (row, kpair):  { x[2p], x[2p+1], logx[2p], logx[2p+1] }
// PB record (32 B) per (kpair, col):  { w0,w1, l2_0,l2_1, l1_0,l1_1, n2_0,n2_1 }
// Each WMMA B-fragment (v2f) is a contiguous half of a b128 load.

// ---------------------------------------------------------------------------
// Prep 1: w1 = tanh(w_hat)*sigmoid(m_hat); derived log/sign matrices packed
// into PB. The faithful TF-reshape transpose is folded into the store index.
// Clamping logs at -1e30 makes exact-zero factors drive exp(sum)->0 without
// 0*(-inf)=NaN inside the WMMA.
// ---------------------------------------------------------------------------
__global__ void nalu_prep_w_kernel(const float* __restrict__ w_hat,
                                   const float* __restrict__ m_hat,
                                   float* __restrict__ PB) {
    int t = blockIdx.x * blockDim.x + threadIdx.x;   // t in [0, NI*NO)
    int r = t >> 9;
    int c = t & 511;
    float wv = tanhf(w_hat[t]) * (1.0f / (1.0f + expf(-m_hat[t])));
    // w1 as B element (K=r, N=c)
    PB[(((r >> 1) * NO + c) << 3) + (r & 1)] = wv;
    float wsv = fabsf(wv);
    float t2 = 1.0f - 2.0f * wsv;
    float t1 = 1.0f - wsv;
    // derived values as B elements (K=c, N=r)  [transposed]
    int d = (((c >> 1) * NO + r) << 3) + (c & 1);
    PB[d + 2] = fmaxf(logf(fabsf(t2)), NEG_BIG);     // log(0) -> -inf -> clamp
    PB[d + 4] = fmaxf(logf(t1), NEG_BIG);            // t1 >= 0 always
    PB[d + 6] = (t2 < 0.0f) ? 1.0f : 0.0f;
}

// Prep 2: pack {x, logx} pairs into PA records.
__global__ void nalu_prep_x_kernel(const float* __restrict__ x,
                                   float* __restrict__ PA) {
    int t = blockIdx.x * blockDim.x + threadIdx.x;   // t in [0, NB*NI)
    int b = t >> 9;
    int i = t & 511;
    float xv = x[t];
    int d = ((b * (NI >> 1) + (i >> 1)) << 2) + (i & 1);
    PA[d]     = xv;
    PA[d + 2] = logf(fmaxf(fabsf(xv), EPS));
}

// ---------------------------------------------------------------------------
// Main: one wave per 32x16 output tile (two 16x16 WMMA tiles sharing the B
// fragments). Per k-step: 2x A b128 + 2x B b128, 10 f32 WMMAs.
//   accA = x@w1, accM = logx@w1, accL = n@L2 + z@L1, accP = n@N2.
// Uniform control flow keeps EXEC all-ones as WMMA requires.
// ---------------------------------------------------------------------------
__global__ void nalu_main_kernel(const float* __restrict__ PA,
                                 const float* __restrict__ g,
                                 const float* __restrict__ PB,
                                 float* __restrict__ out) {
    const int lane = threadIdx.x & 31;
    const int wave = threadIdx.x >> 5;
    const int tile = blockIdx.x * 8 + wave;          // 1024 tiles: 32 x 32
    const int m0 = (tile >> 5) << 5;                 // 32-row supertile
    const int n0 = (tile & 31) << 4;                 // 16-col tile
    const int half = lane >> 4;
    const int l15 = lane & 15;
    const int bcol = n0 + l15;

    const v4f* pa0 = (const v4f*)PA + (m0 + l15) * (NI >> 1) + half;
    const v4f* pa1 = pa0 + 16 * (NI >> 1);           // rows m0+16 .. m0+31
    const v4f* pb  = (const v4f*)PB + ((half * NO + bcol) << 1);

    v8f accA0 = {}, accM0 = {}, accL0 = {}, accP0 = {};
    v8f accA1 = {}, accM1 = {}, accL1 = {}, accP1 = {};

#pragma unroll 4
    for (int k = 0; k < NI; k += 4) {
        // ---- A records: {x pair, logx pair} ----
        v4f a0 = pa0[0];  pa0 += 2;
        v4f a1 = pa1[0];  pa1 += 2;
        // ---- B record: two b128 from one base ----
        v4f bq0 = pb[0];                             // {w pair, l2 pair}
        v4f bq1 = pb[1];                             // {l1 pair, n2 pair}
        pb += 4 * NO;                                // advance 2 k-pairs

        v2f ax0 = __builtin_shufflevector(a0, a0, 0, 1);
        v2f al0 = __builtin_shufflevector(a0, a0, 2, 3);
        v2f ax1 = __builtin_shufflevector(a1, a1, 0, 1);
        v2f al1 = __builtin_shufflevector(a1, a1, 2, 3);
        v2f bw  = __builtin_shufflevector(bq0, bq0, 0, 1);
        v2f b2  = __builtin_shufflevector(bq0, bq0, 2, 3);
        v2f b1  = __builtin_shufflevector(bq1, bq1, 0, 1);
        v2f bn  = __builtin_shufflevector(bq1, bq1, 2, 3);

        v2f an0, az0, an1, az1;
        an0.x = (ax0.x < 0.0f) ? 1.0f : 0.0f;
        an0.y = (ax0.y < 0.0f) ? 1.0f : 0.0f;
        az0.x = (ax0.x == 0.0f) ? 1.0f : 0.0f;
        az0.y = (ax0.y == 0.0f) ? 1.0f : 0.0f;
        an1.x = (ax1.x < 0.0f) ? 1.0f : 0.0f;
        an1.y = (ax1.y < 0.0f) ? 1.0f : 0.0f;
        az1.x = (ax1.x == 0.0f) ? 1.0f : 0.0f;
        az1.y = (ax1.y == 0.0f) ? 1.0f : 0.0f;

        accA0 = __builtin_amdgcn_wmma_f32_16x16x4_f32(false, ax0, false, bw,
                                                      (short)0, accA0, false, false);
        accA1 = __builtin_amdgcn_wmma_f32_16x16x4_f32(false, ax1, false, bw,
                                                      (short)0, accA1, false, false);
        accM0 = __builtin_amdgcn_wmma_f32_16x16x4_f32(false, al0, false, bw,
                                                      (short)0, accM0, false, false);
        accM1 = __builtin_amdgcn_wmma_f32_16x16x4_f32(false, al1, false, bw,
                                                      (short)0, accM1, false, false);
        accL0 = __builtin_amdgcn_wmma_f32_16x16x4_f32(false, an0, false, b2,
                                                      (short)0, accL0, false, false);
        accL1 = __builtin_amdgcn_wmma_f32_16x16x4_f32(false, an1, false, b2,
                                                      (short)0, accL1, false, false);
        accL0 = __builtin_amdgcn_wmma_f32_16x16x4_f32(false, az0, false, b1,
                                                      (short)0, accL0, false, false);
        accL1 = __builtin_amdgcn_wmma_f32_16x16x4_f32(false, az1, false, b1,
                                                      (short)0, accL1, false, false);
        accP0 = __builtin_amdgcn_wmma_f32_16x16x4_f32(false, an0, false, bn,
                                                      (short)0, accP0, false, false);
        accP1 = __builtin_amdgcn_wmma_f32_16x16x4_f32(false, an1, false, bn,
                                                      (short)0, accP1, false, false);
    }

    // ---- epilogue ----
    const float g1 = 1.0f / (1.0f + expf(-g[bcol]));
    const float gm = 1.0f - g1;
#pragma unroll
    for (int v = 0; v < 8; ++v) {
        const int row0 = m0 + v + 8 * half;          // C/D layout
        {
            const float m1 = expf(fminf(accM0[v], LOG_CLIP));
            const int negc = (int)(accP0[v] + 0.5f);
            float ms = ((negc & 1) ? -1.0f : 1.0f) * expf(accL0[v]);
            ms = fminf(fmaxf(ms, -1.0f), 1.0f);
            out[row0 * NO + bcol] = g1 * accA0[v] + gm * m1 * ms;
        }
        {
            const float m1 = expf(fminf(accM1[v], LOG_CLIP));
            const int negc = (int)(accP1[v] + 0.5f);
            float ms = ((negc & 1) ? -1.0f : 1.0f) * expf(accL1[v]);
            ms = fminf(fmaxf(ms, -1.0f), 1.0f);
            out[(row0 + 16) * NO + bcol] = g1 * accA1[v] + gm * m1 * ms;
        }
    }
}

extern "C" void kernel_launch(void* const* d_in, const int* in_sizes, int n_in,
                              void* d_out, int out_size, void* d_ws, size_t ws_size,
                              hipStream_t stream) {
    const float* x     = (const float*)d_in[0];   // [1024,512]
    const float* w_hat = (const float*)d_in[1];   // [512,512]
    const float* m_hat = (const float*)d_in[2];   // [512,512]
    const float* g     = (const float*)d_in[3];   // [512]
    float* out = (float*)d_out;                   // [1024,512]

    float* wsf = (float*)d_ws;                    // 8 MB total
    float* PB  = wsf;                             // (NI/2)*NO*8 = 1M floats
    float* PA  = wsf + (NI / 2) * NO * 8;         // NB*(NI/2)*4 = 1M floats

    nalu_prep_w_kernel<<<(NI * NO) / 256, 256, 0, stream>>>(w_hat, m_hat, PB);
    nalu_prep_x_kernel<<<(NB * NI) / 256, 256, 0, stream>>>(x, PA);

    // 1024 supertiles (32x16 out each), one wave32 per tile, 8 waves/block
    nalu_main_kernel<<<128, 256, 0, stream>>>(PA, g, PB, out);
}